// EdgeAttention_25744033972452
// MI455X (gfx1250) — compile-verified
//
#include <hip/hip_runtime.h>
#include <math.h>

typedef __attribute__((ext_vector_type(2))) float v2f;
typedef __attribute__((ext_vector_type(8))) float v8f;

#define IN_FEAT   256
#define KSTEPS    (IN_FEAT / 4)          // 64 WMMA K-steps
#define BIMG_SZ   (KSTEPS * 32 * 2)      // per-lane B image: 4096 floats = 16 KB

// ---------------------------------------------------------------------------
// Kernel 1: in-degree of col[] via u32 atomics (deterministic).
// ---------------------------------------------------------------------------
__global__ void deg_kernel(const int* __restrict__ col, unsigned* __restrict__ deg, int E) {
    int i = blockIdx.x * blockDim.x + threadIdx.x;
    if (i < E) {
        atomicAdd(&deg[col[i]], 1u);
    }
}

// ---------------------------------------------------------------------------
// Kernel 2: dis[n] = deg^-0.5  (deg==0 -> +inf, matching jnp pow(-0.5)).
// ---------------------------------------------------------------------------
__global__ void dis_kernel(const unsigned* __restrict__ deg, float* __restrict__ dis, int N) {
    int i = blockIdx.x * blockDim.x + threadIdx.x;
    if (i < N) {
        float d = (float)deg[i];
        dis[i] = 1.0f / sqrtf(d);
    }
}

// ---------------------------------------------------------------------------
// Kernel 3: per-node gates via V_WMMA_F32_16X16X4_F32 (one wave32 / 16 nodes).
//   D[16x16] += A[16x4] * B[4x16], 64 K-steps sweep IN_FEAT=256.
//   A = x-tile: lane m holds {K=k0+2*kh, +1} as a contiguous float2 (ISA 7.12.2).
//   B: col 0 = p_w, col 1 = q_w, cols 2..15 = 0.  The per-lane B image for all
//   64 K-steps is prebuilt in LDS -> hot loop has ZERO exec-mask branching:
//   global_load_b64 (A) + ds_load_b64 (B) + v_wmma.
//   D: lane 0/16 hold p_val of nodes 0..7/8..15; lanes 1/17 hold q_val.
// ---------------------------------------------------------------------------
__global__ void gate_wmma_kernel(const float* __restrict__ x,
                                 const float* __restrict__ pw,
                                 const float* __restrict__ qw,
                                 const float* __restrict__ pb,
                                 const float* __restrict__ qb,
                                 float* __restrict__ pval,
                                 float* __restrict__ qval,
                                 int N, int nTiles) {
    __shared__ float bimg[BIMG_SZ];      // [kstep][lane][2]

    // Cooperative build of the B VGPR image (all waves, incl. tail-inactive).
    for (int idx = threadIdx.x; idx < BIMG_SZ; idx += blockDim.x) {
        int j  = idx & 1;                // float within the pair
        int ln = (idx >> 1) & 31;        // target lane
        int it = idx >> 6;               // K-step 0..63
        int n  = ln & 15;                // B column
        int kh = ln >> 4;                // K half-select
        int k  = it * 4 + 2 * kh + j;    // feature index
        float v = 0.f;
        if (n == 0)      v = pw[k];
        else if (n == 1) v = qw[k];
        bimg[idx] = v;
    }
    __syncthreads();

    int wave = (blockIdx.x * blockDim.x + threadIdx.x) >> 5;   // global wave id
    int lane = threadIdx.x & 31;

    if (wave < nTiles) {                                       // wave-uniform
        int node0 = wave * 16;
        int m  = lane & 15;              // A row (M) / D column (N)
        int kh = lane >> 4;              // K half-select

        int node = node0 + m;
        if (node >= N) node = N - 1;     // tail clamp (N % 16 == 0 here)
        const float* xrow = x + (size_t)node * IN_FEAT + 2 * kh;
        const float* brow = &bimg[lane * 2];

        v8f c = {0.f, 0.f, 0.f, 0.f, 0.f, 0.f, 0.f, 0.f};

#pragma unroll 8
        for (int it = 0; it < KSTEPS; ++it) {
            v2f a = *(const v2f*)(xrow + it * 4);
            v2f b = *(const v2f*)(brow + it * 64);
            // 8 args: (neg_a, A, neg_b, B, c_mod, C, reuse_a, reuse_b)
            c = __builtin_amdgcn_wmma_f32_16x16x4_f32(
                    false, a, false, b, (short)0, c, false, false);
        }

        // Writeback: column 0 lanes -> p, column 1 lanes -> q.
        if (m == 0) {
            float bias = pb[0];
            int base = node0 + 8 * kh;
#pragma unroll
            for (int i = 0; i < 8; ++i) {
                if (base + i < N) {
                    float v = c[i] + bias;
                    pval[base + i] = v > 0.f ? v : 0.f;
                }
            }
        } else if (m == 1) {
            float bias = qb[0];
            int base = node0 + 8 * kh;
#pragma unroll
            for (int i = 0; i < 8; ++i) {
                if (base + i < N) {
                    float v = c[i] + bias;
                    qval[base + i] = v > 0.f ? v : 0.f;
                }
            }
        }
    }
}

// ---------------------------------------------------------------------------
// Kernel 4: edge pass.
//   out[e] = dis[row]*ea*dis[col]*p[col] + q[row]*ea
// Streams 16 B/edge; gathers hit 3 x 400 KB tables resident in the 192 MB L2.
// ---------------------------------------------------------------------------
__global__ void edge_kernel(const int* __restrict__ row,
                            const int* __restrict__ col,
                            const float* __restrict__ ea,
                            const float* __restrict__ dis,
                            const float* __restrict__ pval,
                            const float* __restrict__ qval,
                            float* __restrict__ out, int E) {
    int i = blockIdx.x * blockDim.x + threadIdx.x;
    if (i < E) {
        int r = row[i];
        int c = col[i];
        float a = ea[i];
        out[i] = dis[r] * a * dis[c] * pval[c] + qval[r] * a;
    }
}

// ---------------------------------------------------------------------------
extern "C" void kernel_launch(void* const* d_in, const int* in_sizes, int n_in,
                              void* d_out, int out_size, void* d_ws, size_t ws_size,
                              hipStream_t stream) {
    const float* x  = (const float*)d_in[0];
    const int*   ei = (const int*)d_in[1];   // [2, E]: row = ei, col = ei + E (int32: JAX x64 off)
    const float* ea = (const float*)d_in[2];
    const float* pw = (const float*)d_in[3];
    const float* pb = (const float*)d_in[4];
    const float* qw = (const float*)d_in[5];
    const float* qb = (const float*)d_in[6];

    const int N = in_sizes[0] / IN_FEAT;     // 100000
    const int E = in_sizes[2];               // 3200000

    const int* row = ei;
    const int* col = ei + E;

    // Workspace layout: deg(u32)[N] | dis[N] | pval[N] | qval[N]
    unsigned* deg  = (unsigned*)d_ws;
    float*    dis  = (float*)d_ws + N;
    float*    pval = (float*)d_ws + 2 * (size_t)N;
    float*    qval = (float*)d_ws + 3 * (size_t)N;

    hipMemsetAsync(deg, 0, (size_t)N * sizeof(unsigned), stream);

    deg_kernel<<<(E + 255) / 256, 256, 0, stream>>>(col, deg, E);
    dis_kernel<<<(N + 255) / 256, 256, 0, stream>>>(deg, dis, N);

    const int nTiles = (N + 15) / 16;        // 6250 tiles, one wave32 each
    const int wavesPerBlock = 8;             // 256 threads
    gate_wmma_kernel<<<(nTiles + wavesPerBlock - 1) / wavesPerBlock, 32 * wavesPerBlock, 0, stream>>>(
        x, pw, qw, pb, qb, pval, qval, N, nTiles);

    edge_kernel<<<(E + 255) / 256, 256, 0, stream>>>(row, col, ea, dis, pval, qval,
                                                     (float*)d_out, E);
}